// SparseDenseMatMulCPU_37443524887286
// MI455X (gfx1250) — compile-verified
//
#include <hip/hip_runtime.h>

// ---------------------------------------------------------------------------
// COO SpMM: out[M,N] = segment_sum(vals[:,None] * B[cols], rows)
//   M = K = 16384, NNZ = 1048576, N = 256, all fp32 (indices int64).
//
// Memory-bound / atomic-bound problem: B (16MB) and out (16MB) are L2
// resident (192MB L2). Compute (0.54 GFLOP) is free, so we run the
// per-nonzero scaling on the WMMA pipe:
//   contrib(16x256) = diag(v_16) x G(16x256),  G = gathered B rows,
// decomposed into 4 slabs A_b (16x4) so that sum_b A_b x B_b == diag(v) x G,
// evaluated per 16-column tile with V_WMMA_F32_16X16X4_F32.
// ---------------------------------------------------------------------------

typedef __attribute__((ext_vector_type(2))) float v2f;
typedef __attribute__((ext_vector_type(4))) float v4f;
typedef __attribute__((ext_vector_type(8))) float v8f;

#define SP_NNZ     1048576
#define SP_NDENSE  256
#define SP_GROUP   16          // nonzeros per wave
#define SP_WAVES   8           // waves per 256-thread block

// ---------------------------------------------------------------------------
__global__ __launch_bounds__(256)
void spmm_zero_out(float* __restrict__ out, int n_elems) {
    int i = blockIdx.x * blockDim.x + threadIdx.x;
    int idx4 = i * 4;
    if (idx4 + 3 < n_elems) {
        v4f z = {0.0f, 0.0f, 0.0f, 0.0f};
        *(v4f*)(out + idx4) = z;
    } else {
        for (int k = idx4; k < n_elems; ++k) out[k] = 0.0f;
    }
}

// ---------------------------------------------------------------------------
__global__ __launch_bounds__(256)
void spmm_coo_wmma(const float*     __restrict__ vals,
                   const long long* __restrict__ rows,   // A_indices[0, :]
                   const long long* __restrict__ cols,   // A_indices[1, :]
                   const float*     __restrict__ B,      // [K_COLS, N]
                   float*           __restrict__ out) {  // [M_ROWS, N]
    const int lane = threadIdx.x & 31;
    const int wave = threadIdx.x >> 5;
    const int g    = blockIdx.x * SP_WAVES + wave;       // group of 16 nnz
    const int base = g * SP_GROUP;

    const int m  = lane & 15;        // which nonzero of the group this lane owns
    const int hi = lane >> 4;        // lane half (selects K pair / M+8)

    // Per-lane copy of nonzero m's triple (coalesced loads).
    const float val_m = vals[base + m];
    const int   row_m = (int)rows[base + m];
    const int   col_m = (int)cols[base + m];

    // Scatter rows: accumulator VGPR r (r=0..7) holds output row srow[r].
    int srow[8];
#pragma unroll
    for (int r = 0; r < 8; ++r)
        srow[r] = __shfl(row_m, r + 8 * hi, 32);

    const int k0 = hi ? 2 : 0;       // this lane half's K offset within a slab

    // A operands for the 4 slabs: A_b[M=m, K] = val_m iff m == 4b+K.
    // ISA 16x4 f32 A layout: lane m<16 -> (K=0 in v.x, K=1 in v.y);
    //                        lane m+16 -> (K=2, K=3).
    v2f A[4];
#pragma unroll
    for (int b = 0; b < 4; ++b) {
        A[b].x = (m == 4 * b + k0)     ? val_m : 0.0f;
        A[b].y = (m == 4 * b + k0 + 1) ? val_m : 0.0f;
    }

    // Column indices feeding B_b[k, :] = B[ c_{4b+k}, tile ].
    int c0[4], c1[4];
#pragma unroll
    for (int b = 0; b < 4; ++b) {
        c0[b] = __shfl(col_m, 4 * b + k0,     32);
        c1[b] = __shfl(col_m, 4 * b + k0 + 1, 32);
    }

    // 16 column tiles of 16 cover N = 256.
#pragma unroll 4
    for (int t = 0; t < SP_NDENSE / 16; ++t) {
        const int n = t * 16 + m;    // column handled by this lane (N = lane%16)

        v8f acc = {};                // C starts at zero, accumulates over slabs
#pragma unroll
        for (int b = 0; b < 4; ++b) {
            // ISA 4x16 f32 B layout: lane n<16 -> rows K=0 (.x), K=1 (.y);
            //                        lane n+16 -> rows K=2, K=3.
            v2f Bb;
            Bb.x = B[(long)c0[b] * SP_NDENSE + n];
            Bb.y = B[(long)c1[b] * SP_NDENSE + n];
            // D = A_b x B_b + acc   (8 args: neg_a, A, neg_b, B, c_mod, C, reuse_a, reuse_b)
            acc = __builtin_amdgcn_wmma_f32_16x16x4_f32(
                false, A[b], false, Bb, (short)0, acc, false, false);
        }

        // Scatter-add: VGPR r holds contrib[row srow[r], col n].
        // Lanes 0..15 of a fixed r touch 64B contiguous output -> coalesced atomics.
#pragma unroll
        for (int r = 0; r < 8; ++r) {
            atomicAdd(out + (long)srow[r] * SP_NDENSE + n, acc[r]);
        }
    }
}

// ---------------------------------------------------------------------------
extern "C" void kernel_launch(void* const* d_in, const int* in_sizes, int n_in,
                              void* d_out, int out_size, void* d_ws, size_t ws_size,
                              hipStream_t stream) {
    const float*     vals = (const float*)d_in[0];       // A_values  [NNZ] f32
    const long long* idx  = (const long long*)d_in[1];   // A_indices [2, NNZ] i64
    const float*     B    = (const float*)d_in[2];       // B [K, N] f32
    float*           out  = (float*)d_out;               // [M, N] f32

    const long long* rows = idx;                         // A_indices[0, :]
    const long long* cols = idx + SP_NNZ;                // A_indices[1, :]

    // Zero the (poisoned) output, then scatter-accumulate.
    int n4 = (out_size + 3) / 4;
    spmm_zero_out<<<(n4 + 255) / 256, 256, 0, stream>>>(out, out_size);

    const int groups = SP_NNZ / SP_GROUP;                // 65536 waves
    spmm_coo_wmma<<<groups / SP_WAVES, 256, 0, stream>>>(vals, rows, cols, B, out);
}